// lsgm_53437983096982
// MI455X (gfx1250) — compile-verified
//
#include <hip/hip_runtime.h>
#include <hip/hip_bf16.h>
#include <cstdint>

#define T_STEPS 256
#define BATCH   32
#define K_STATES 128
#define SMALLF  1.17549435e-38f

#define CHUNK_ROWS   32
#define CHUNK_FLOATS (CHUNK_ROWS * K_STATES)   // 4096 floats = 16 KB
#define NCHUNKS      4                          // chunks per timestep (4*32 = 128 rows)

typedef float v2f __attribute__((ext_vector_type(2)));
typedef float v8f __attribute__((ext_vector_type(8)));
typedef int   v4i __attribute__((vector_size(16)));

// ---------------- async copy helpers (CDNA5 async-to-LDS path) ----------------
__device__ __forceinline__ void async_wait0() {
#if __has_builtin(__builtin_amdgcn_s_wait_asynccnt)
  __builtin_amdgcn_s_wait_asynccnt(0);
#else
  asm volatile("s_wait_asynccnt 0" ::: "memory");
#endif
}

__device__ __forceinline__ void async_copy16(const float* g, float* l) {
#if __has_builtin(__builtin_amdgcn_global_load_async_to_lds_b128)
  typedef __attribute__((address_space(1))) v4i GT;   // global src
  typedef __attribute__((address_space(3))) v4i LT;   // LDS dst
  __builtin_amdgcn_global_load_async_to_lds_b128((GT*)g, (LT*)l, 0, 0);
#else
  // fallback: synchronous 16B copy (still correct, just not overlapped)
  *(float4*)l = *(const float4*)g;
#endif
}

// ---------------- wave32 reductions ----------------
__device__ __forceinline__ float wave_max(float x) {
#pragma unroll
  for (int off = 16; off > 0; off >>= 1) x = fmaxf(x, __shfl_xor(x, off, 32));
  return x;
}
__device__ __forceinline__ float wave_sum(float x) {
#pragma unroll
  for (int off = 16; off > 0; off >>= 1) x += __shfl_xor(x, off, 32);
  return x;
}

// ---------------- kernel 1: forward recursion, one workgroup per batch ----------------
__global__ __launch_bounds__(128)
void hmm_forward_kernel(const float* __restrict__ trans,   // [T,B,K,K] probabilities
                        const float* __restrict__ emis,    // [T,B,K]   log-emissions
                        const float* __restrict__ start,   // [1,K]
                        float* __restrict__ out) {         // [T*B] diffs ++ [T,B,K] fm
  __shared__ __align__(16) float tbuf[2][CHUNK_FLOATS];    // double-buffered trans chunks
  __shared__ float v_lds[K_STATES];
  __shared__ float f_lds[K_STATES];
  __shared__ float red[8];

  const int tid  = threadIdx.x;     // 0..127
  const int lane = tid & 31;
  const int wid  = tid >> 5;        // 0..3
  const int b    = blockIdx.x;      // batch index

  float* fm = out + (size_t)T_STEPS * BATCH;   // forward_matrix region

  auto chunk_ptr = [&](int q) -> const float* {
    const int tt = q >> 2, c = q & 3;          // step t uses transitions[t-1]; tt = t-1
    return trans + ((size_t)(tt * BATCH + b) * (K_STATES * K_STATES)
                    + (size_t)c * CHUNK_FLOATS);
  };
  auto issue_chunk = [&](int q) {
    const float* src = chunk_ptr(q);
    float* dst = &tbuf[q & 1][0];
#pragma unroll
    for (int i = 0; i < 8; ++i) {              // 128 thr * 8 * 16B = 16 KB
      const int off = (i * 128 + tid) * 4;     // float offset, 16B aligned
      async_copy16(src + off, dst + off);
    }
  };

  const int QTOT = (T_STEPS - 1) * NCHUNKS;
  issue_chunk(0);                               // prefetch while we do t=0

  // ---- t = 0 : f0 = log(start + tiny) + e0 ----
  {
    const float e0 = emis[(size_t)b * K_STATES + tid];
    const float f0 = __logf(start[tid] + SMALLF) + e0;
    f_lds[tid] = f0;
    fm[(size_t)b * K_STATES + tid] = f0;
  }

  int q = 0;
  for (int t = 1; t < T_STEPS; ++t) {
    __syncthreads();                            // f_lds from previous step visible
    // c = max_k f_prev ; v = exp(f_prev - c) ; sv = tiny * sum(v)
    const float f = f_lds[tid];
    const float wm = wave_max(f);
    if (lane == 0) red[wid] = wm;
    __syncthreads();
    const float cmax = fmaxf(fmaxf(red[0], red[1]), fmaxf(red[2], red[3]));
    const float v = __expf(f - cmax);
    v_lds[tid] = v;
    const float ws = wave_sum(v);
    if (lane == 0) red[4 + wid] = ws;
    __syncthreads();
    const float sv = (red[4] + red[5] + red[6] + red[7]) * SMALLF;

    // Each wave owns 32 consecutive k' columns: two 16-wide WMMA tiles.
    v8f acc0 = {}; v8f acc1 = {};
    const int base  = wid * 32;
    const int col   = lane & 15;
    const int ksel0 = (lane < 16) ? 0 : 2;      // B/A VGPR0 holds K=0 (lo lanes) / K=2 (hi)
    const int ksel1 = ksel0 + 1;                // B/A VGPR1 holds K=1 / K=3

    for (int c = 0; c < NCHUNKS; ++c, ++q) {
      async_wait0();                            // chunk q resident in tbuf[q&1]
      __syncthreads();
      if (q + 1 < QTOT) issue_chunk(q + 1);     // overlap next chunk's copy
      const float* tb = &tbuf[q & 1][0];
#pragma unroll
      for (int kk = 0; kk < CHUNK_ROWS; kk += 4) {
        v2f a, b0, b1;
        a.x  = v_lds[(c * CHUNK_ROWS) + kk + ksel0];         // A: v broadcast, all M rows equal
        a.y  = v_lds[(c * CHUNK_ROWS) + kk + ksel1];
        b0.x = tb[(kk + ksel0) * K_STATES + base + col];      // B: 4x16 tile of trans (row-major)
        b0.y = tb[(kk + ksel1) * K_STATES + base + col];
        b1.x = tb[(kk + ksel0) * K_STATES + base + 16 + col];
        b1.y = tb[(kk + ksel1) * K_STATES + base + 16 + col];
        acc0 = __builtin_amdgcn_wmma_f32_16x16x4_f32(false, a, false, b0,
                                                     (short)0, acc0, false, false);
        acc1 = __builtin_amdgcn_wmma_f32_16x16x4_f32(false, a, false, b1,
                                                     (short)0, acc1, false, false);
      }
    }
    // All D rows identical; lane's element 0 covers N = lane&15 of its tile.
    const float s  = ((lane < 16) ? acc0[0] : acc1[0]) + sv;
    const int   kp = base + lane;
    const float e  = emis[((size_t)t * BATCH + b) * K_STATES + kp];
    const float fn = __logf(s) + cmax + e;
    f_lds[kp] = fn;                             // old f_lds fully consumed pre chunk-barriers
    fm[((size_t)t * BATCH + b) * K_STATES + kp] = fn;
  }
}

// ---------------- kernel 2: per-(t,b) prefix LSE and diff ----------------
__global__ __launch_bounds__(32)
void hmm_prefix_kernel(const float* __restrict__ fm,   // [T,B,K]
                       float* __restrict__ out1) {     // [T,B]
  const int idx  = blockIdx.x;                 // t*B + b
  const int t    = idx / BATCH;
  const int b    = idx % BATCH;
  const int lane = threadIdx.x;

  auto lse_row = [&](int tt) -> float {
    const float* row = fm + ((size_t)tt * BATCH + b) * K_STATES;
    float x[4]; float m = -INFINITY;
#pragma unroll
    for (int j = 0; j < 4; ++j) { x[j] = row[lane + 32 * j]; m = fmaxf(m, x[j]); }
    m = wave_max(m);
    float s = 0.f;
#pragma unroll
    for (int j = 0; j < 4; ++j) s += __expf(x[j] - m);
    s = wave_sum(s);
    return __logf(s) + m;
  };

  const float cur  = lse_row(t);
  const float prev = (t > 0) ? lse_row(t - 1) : 0.0f;
  if (lane == 0) out1[idx] = cur - prev;
}

extern "C" void kernel_launch(void* const* d_in, const int* in_sizes, int n_in,
                              void* d_out, int out_size, void* d_ws, size_t ws_size,
                              hipStream_t stream) {
  (void)in_sizes; (void)n_in; (void)d_ws; (void)ws_size; (void)out_size;
  // setup_inputs order: sequences (unused), transitions, emissions, start_transitions
  const float* trans = (const float*)d_in[1];
  const float* emis  = (const float*)d_in[2];
  const float* start = (const float*)d_in[3];
  float* out = (float*)d_out;

  hmm_forward_kernel<<<BATCH, 128, 0, stream>>>(trans, emis, start, out);
  hmm_prefix_kernel<<<T_STEPS * BATCH, 32, 0, stream>>>(
      out + (size_t)T_STEPS * BATCH, out);
}